// BaselineSparseAutoencoder_54468775247877
// MI455X (gfx1250) — compile-verified
//
#include <hip/hip_runtime.h>

// Sparse autoencoder forward for MI455X (gfx1250, wave32).
//  K1: f32 WMMA tiled GEMM, double-buffered LDS fed by async global->LDS copies
//  K2: per-row radix-select top-32 in LDS, zero rest (bandwidth-bound)
//  K3: sparse decode recon = h_sparse @ Wd^T + bd (0.5 GFLOP, Wd L2-resident)

typedef __attribute__((ext_vector_type(2))) float v2f;
typedef __attribute__((ext_vector_type(4))) float v4f;
typedef __attribute__((ext_vector_type(8))) float v8f;
typedef __attribute__((ext_vector_type(4))) int   v4i;

// Address-space-qualified int4 pointers for the async-copy builtin
// (param 0 is AS(1) int4*, param 1 is AS(3) int4* per hipcc's diagnostic).
typedef __attribute__((address_space(1))) v4i gas_v4i;
typedef __attribute__((address_space(3))) v4i las_v4i;

#define BATCH   8192
#define IN_DIM  1024
#define HID     16384
#define TOPK    32

// ---- K1 tiling: block 64(M) x 128(N), chunks of BK=32 along K ----
#define BM      64
#define BN      128
#define BK      32
#define LDK     36                  // BK + 4 pad: rows stay 16B aligned, banks shift
#define A_TILE  (BM * LDK)          // 2304 floats per buffer
#define B_TILE  (BN * LDK)          // 4608 floats per buffer
#define NCHUNK  (IN_DIM / BK)       // 32
#define LDS_BYTES ((2 * A_TILE + 2 * B_TILE) * 4)   // 55296 B dynamic LDS

// ---- async global->LDS plumbing (ASYNCcnt-tracked) ----
#if defined(__has_builtin)
#  if __has_builtin(__builtin_amdgcn_global_load_async_to_lds_b128)
#    define HAVE_ASYNC 1
#  endif
#endif

#ifdef HAVE_ASYNC
__device__ __forceinline__ void async_copy16(const float* g, float* l) {
    __builtin_amdgcn_global_load_async_to_lds_b128(
        (gas_v4i*)g, (las_v4i*)l, 0, 0);
}
#  if __has_builtin(__builtin_amdgcn_s_wait_asynccnt)
#    define WAIT_ASYNC(n) __builtin_amdgcn_s_wait_asynccnt(n)
#  else
#    define WAIT_ASYNC(n) asm volatile("s_wait_asynccnt %0" ::"n"(n) : "memory")
#  endif
#else
// Fallback: synchronous vector copy through VGPRs.
__device__ __forceinline__ void async_copy16(const float* g, float* l) {
    *(v4f*)l = *(const v4f*)g;
}
#  define WAIT_ASYNC(n) ((void)0)
#endif

__device__ __forceinline__ v8f wmma4(v2f a, v2f b, v8f c) {
    return __builtin_amdgcn_wmma_f32_16x16x4_f32(
        false, a, false, b, (short)0, c, false, false);
}

__global__ __launch_bounds__(256) void enc_gemm_kernel(
    const float* __restrict__ x,   // [BATCH, IN_DIM]
    const float* __restrict__ We,  // [HID, IN_DIM]
    const float* __restrict__ be,  // [HID]
    float* __restrict__ h)         // [BATCH, HID]
{
    extern __shared__ float lds[];
    float* As = lds;                 // [2][BM][LDK]
    float* Bs = lds + 2 * A_TILE;    // [2][BN][LDK]

    const int tid  = threadIdx.x;
    const int lane = tid & 31;
    const int wave = tid >> 5;       // 0..7
    const int wm   = wave >> 2;      // 0..1 : 32-row M slab
    const int wn   = wave & 3;       // 0..3 : 32-col N slab

    const int block_m = blockIdx.y * BM;
    const int block_n = blockIdx.x * BN;

    const int ml = lane & 15;        // M (or N) within a 16-tile
    const int kl = (lane >> 4) * 2;  // K pair base per ISA A/B f32 layout

    // acc[mi][ni]: wave computes a 32x32 C tile as 4 16x16 WMMA tiles.
    const float bias0 = be[block_n + wn * 32 + ml];
    const float bias1 = be[block_n + wn * 32 + 16 + ml];
    v8f acc[2][2];
#pragma unroll
    for (int i = 0; i < 8; ++i) {
        acc[0][0][i] = bias0; acc[1][0][i] = bias0;
        acc[0][1][i] = bias1; acc[1][1][i] = bias1;
    }

    // Issue one chunk's tile copies: 6 x b128 per thread (2 for A, 4 for B).
#define STAGE(buf, k0)                                                        \
    do {                                                                      \
        _Pragma("unroll")                                                     \
        for (int i = 0; i < 2; ++i) {                                         \
            const int idx = tid + i * 256;                                    \
            const int r = idx >> 3, c = (idx & 7) * 4;                        \
            async_copy16(x + (size_t)(block_m + r) * IN_DIM + (k0) + c,       \
                         As + (buf)*A_TILE + r * LDK + c);                    \
        }                                                                     \
        _Pragma("unroll")                                                     \
        for (int i = 0; i < 4; ++i) {                                         \
            const int idx = tid + i * 256;                                    \
            const int r = idx >> 3, c = (idx & 7) * 4;                        \
            async_copy16(We + (size_t)(block_n + r) * IN_DIM + (k0) + c,      \
                         Bs + (buf)*B_TILE + r * LDK + c);                    \
        }                                                                     \
    } while (0)

    STAGE(0, 0);

    for (int ch = 0; ch < NCHUNK; ++ch) {
        const int buf = ch & 1;
        if (ch + 1 < NCHUNK) {
            STAGE(buf ^ 1, (ch + 1) * BK);  // next buffer fills in background
            WAIT_ASYNC(6);                  // my 6 copies for *this* buf done
        } else {
            WAIT_ASYNC(0);
        }
        __syncthreads();                    // everyone's copies for buf done

        const float* Ab = As + buf * A_TILE + (wm * 32 + ml) * LDK;
        const float* Bb = Bs + buf * B_TILE + (wn * 32 + ml) * LDK;

        // Software-pipelined fragments: load k-step kk+4 while WMMAing kk.
        v2f a0 = *(const v2f*)(Ab + kl);
        v2f a1 = *(const v2f*)(Ab + 16 * LDK + kl);
        v2f b0 = *(const v2f*)(Bb + kl);
        v2f b1 = *(const v2f*)(Bb + 16 * LDK + kl);
#pragma unroll
        for (int kk = 0; kk < BK; kk += 4) {
            v2f na0 = a0, na1 = a1, nb0 = b0, nb1 = b1;
            if (kk + 4 < BK) {
                na0 = *(const v2f*)(Ab + kk + 4 + kl);
                na1 = *(const v2f*)(Ab + 16 * LDK + kk + 4 + kl);
                nb0 = *(const v2f*)(Bb + kk + 4 + kl);
                nb1 = *(const v2f*)(Bb + 16 * LDK + kk + 4 + kl);
            }
            acc[0][0] = wmma4(a0, b0, acc[0][0]);
            acc[0][1] = wmma4(a0, b1, acc[0][1]);
            acc[1][0] = wmma4(a1, b0, acc[1][0]);
            acc[1][1] = wmma4(a1, b1, acc[1][1]);
            a0 = na0; a1 = na1; b0 = nb0; b1 = nb1;
        }
        __syncthreads();                    // buf free for stage ch+2
    }
#undef STAGE

    // Store: VGPR r holds M = (lane/16)*8 + r within each 16x16 tile.
    const int m_hi = (lane >> 4) * 8;
#pragma unroll
    for (int ni = 0; ni < 2; ++ni) {
        const int n_out = block_n + wn * 32 + ni * 16 + ml;
#pragma unroll
        for (int r = 0; r < 8; ++r) {
            const int m0 = block_m + wm * 32 + m_hi + r;
            h[(size_t)m0 * HID + n_out]        = acc[0][ni][r];
            h[(size_t)(m0 + 16) * HID + n_out] = acc[1][ni][r];
        }
    }
}

// ---- K2: per-row top-32 radix select, zero losers, compact (idx,val) ----
__device__ __forceinline__ unsigned f32_key(float f) {
    unsigned b = __float_as_uint(f);
    return (b & 0x80000000u) ? ~b : (b | 0x80000000u);
}
__device__ __forceinline__ float key_f32(unsigned u) {
    return (u & 0x80000000u) ? __uint_as_float(u & 0x7FFFFFFFu)
                             : __uint_as_float(~u);
}

__global__ __launch_bounds__(256) void topk_kernel(
    float* __restrict__ h,         // [BATCH, HID], zeroed in place
    float* __restrict__ nnz_val,   // [BATCH, TOPK]
    int*   __restrict__ nnz_idx)   // [BATCH, TOPK]
{
    extern __shared__ unsigned keys[];        // HID * 4B = 64 KB dynamic LDS
    __shared__ unsigned hist[16];
    __shared__ unsigned s_prefix, s_k, s_cnt, s_tie;

    const int tid = threadIdx.x;
    const int row = blockIdx.x;
    float* hrow = h + (size_t)row * HID;

    for (int i = tid; i < HID; i += 256) keys[i] = f32_key(hrow[i]);
    if (tid == 0) { s_prefix = 0u; s_k = TOPK; }
    __syncthreads();

    // 8 passes, 4 bits MSB-first: exact key of the 32nd-largest element.
    for (int shift = 28; shift >= 0; shift -= 4) {
        if (tid < 16) hist[tid] = 0u;
        __syncthreads();
        const unsigned prefix = s_prefix;
        const unsigned k      = s_k;
        const unsigned mask   = (shift == 28) ? 0u : (0xFFFFFFFFu << (shift + 4));
        for (int i = tid; i < HID; i += 256) {
            const unsigned u = keys[i];
            if ((u & mask) == prefix) atomicAdd(&hist[(u >> shift) & 15], 1u);
        }
        __syncthreads();
        if (tid == 0) {
            unsigned cum = 0; int b = 15;
            for (; b > 0; --b) {
                const unsigned c = hist[b];
                if (cum + c >= k) break;
                cum += c;
            }
            s_prefix = prefix | ((unsigned)b << shift);
            s_k      = k - cum;     // ties still to take at this prefix level
        }
        __syncthreads();
    }

    if (tid == 0) { s_cnt = 0u; s_tie = 0u; }
    __syncthreads();
    const unsigned thr    = s_prefix;   // key of 32nd-largest
    const unsigned budget = s_k;        // how many ==thr to keep

    for (int i = tid; i < HID; i += 256) {
        const unsigned u = keys[i];
        bool keep = (u > thr);
        if (!keep && u == thr) keep = (atomicAdd(&s_tie, 1u) < budget);
        if (keep) {
            const unsigned p = atomicAdd(&s_cnt, 1u);
            nnz_val[row * TOPK + p] = key_f32(u);
            nnz_idx[row * TOPK + p] = i;
        } else {
            hrow[i] = 0.0f;
        }
    }
}

// ---- K3: sparse decode, recon = sum_t val_t * Wd[:, idx_t] + bd ----
__global__ __launch_bounds__(256) void decode_kernel(
    const float* __restrict__ nnz_val,
    const int*   __restrict__ nnz_idx,
    const float* __restrict__ Wd,   // [IN_DIM, HID]
    const float* __restrict__ bd,   // [IN_DIM]
    float* __restrict__ recon)      // [BATCH, IN_DIM]
{
    __shared__ float sval[TOPK];
    __shared__ int   sidx[TOPK];
    const int tid = threadIdx.x;
    const int row = blockIdx.x;

    if (tid < TOPK) {
        sval[tid] = nnz_val[row * TOPK + tid];
        sidx[tid] = nnz_idx[row * TOPK + tid];
    }
    __syncthreads();

    for (int j = tid; j < IN_DIM; j += 256) {
        float acc = bd[j];
        const float* wrow = Wd + (size_t)j * HID;   // L2-resident (64 MB < 192 MB)
#pragma unroll
        for (int t = 0; t < TOPK; ++t) acc += sval[t] * wrow[sidx[t]];
        recon[(size_t)row * IN_DIM + j] = acc;
    }
}

extern "C" void kernel_launch(void* const* d_in, const int* in_sizes, int n_in,
                              void* d_out, int out_size, void* d_ws, size_t ws_size,
                              hipStream_t stream) {
    const float* x  = (const float*)d_in[0];
    const float* We = (const float*)d_in[1];
    const float* be = (const float*)d_in[2];
    const float* Wd = (const float*)d_in[3];
    const float* bd = (const float*)d_in[4];

    float* h_sparse = (float*)d_out;                          // [BATCH, HID]
    float* recon    = h_sparse + (size_t)BATCH * HID;         // [BATCH, IN_DIM]

    float* nnz_val = (float*)d_ws;                            // 1 MB
    int*   nnz_idx = (int*)((char*)d_ws +
                            (size_t)BATCH * TOPK * sizeof(float)); // 1 MB

    // K1: encoder GEMM (WMMA f32, async double-buffered LDS)
    dim3 gridG(HID / BN, BATCH / BM);
    enc_gemm_kernel<<<gridG, 256, LDS_BYTES, stream>>>(x, We, be, h_sparse);

    // K2: top-k select per row (64 KB dynamic LDS for row keys)
    const size_t shmem = (size_t)HID * sizeof(unsigned);
    topk_kernel<<<BATCH, 256, shmem, stream>>>(h_sparse, nnz_val, nnz_idx);

    // K3: sparse decode
    decode_kernel<<<BATCH, 256, 0, stream>>>(nnz_val, nnz_idx, Wd, bd, recon);
}